// Encoder_TRANSFORMERRECURRENT_6751688589491
// MI455X (gfx1250) — compile-verified
//
#include <hip/hip_runtime.h>

// ---------------- problem constants ----------------
#define BS      8
#define TT      80
#define AA      6
#define NL      4
#define NH      4
#define DH      192
#define DM      768
#define FFD     1024
#define LATENT  256
#define INF_    150          // nj * nf
#define NR      (TT * BS)    // 640 rows for the layer-0 batched GEMM

// ---------------- workspace layout (floats) ----------------
#define OFF_X0  0
#define OFF_Q0  (NR * DM)
#define OFF_K0  (2 * NR * DM)
#define OFF_V0  (3 * NR * DM)
#define OFF_S   (4 * NR * DM)
#define SZ_S    (NL * BS * NH * DH * DH)
#define OFF_Z   (OFF_S + SZ_S)
#define SZ_Z    (NL * BS * NH * DH)

typedef float v2f __attribute__((ext_vector_type(2)));
typedef float v8f __attribute__((ext_vector_type(8)));

// ---------------- zero S/Z state every call ----------------
__global__ void zero_f32(float* __restrict__ p, int n) {
  int i = blockIdx.x * blockDim.x + threadIdx.x;
  int stride = gridDim.x * blockDim.x;
  for (; i < n; i += stride) p[i] = 0.0f;
}

// ---------------- build X0 = [T*bs, DM]: (mu_q+pe0 | sig_q+pe1 | xe+pe2) ----
__global__ __launch_bounds__(256) void build_x0(
    const float* __restrict__ x, const int* __restrict__ y,
    const int* __restrict__ ind_map,
    const float* __restrict__ skel_W, const float* __restrict__ skel_b,
    const float* __restrict__ muQ, const float* __restrict__ sigQ,
    float* __restrict__ X0)
{
  int t = blockIdx.x, b = blockIdx.y, c = threadIdx.x;
  __shared__ float xrow[INF_];
  if (c < INF_) xrow[c] = x[(b * INF_ + c) * TT + t];
  __syncthreads();

  float acc = skel_b[c];
  #pragma unroll 5
  for (int j = 0; j < INF_; ++j)
    acc = fmaf(xrow[j], skel_W[j * LATENT + c], acc);

  int a   = ind_map[t * BS + b];
  int cls = y[b * AA + a];
  float cm = muQ[cls * LATENT + c];
  float cs = sigQ[cls * LATENT + c];

  int start = (t == 0) ? 0 : (t + 2);
  // pe[pos][c] = (c even ? sin : cos)(pos * exp(-(c&~1) * ln(1e4)/LATENT))
  float freq = expf((float)(c & ~1) * (-logf(10000.0f) / (float)LATENT));
  bool odd = (c & 1);
  float p0 = odd ? cosf((float)(start)     * freq) : sinf((float)(start)     * freq);
  float p1 = odd ? cosf((float)(start + 1) * freq) : sinf((float)(start + 1) * freq);
  float p2 = odd ? cosf((float)(start + 2) * freq) : sinf((float)(start + 2) * freq);

  int row = t * BS + b;
  X0[row * DM + c]              = cm  + p0;
  X0[row * DM + LATENT + c]     = cs  + p1;
  X0[row * DM + 2 * LATENT + c] = acc + p2;
}

// ---------------- layer-0 QKV: [640 x 768] @ [768 x 768] via f32 WMMA -------
// Fragment packing per CDNA5 ISA VGPR layout tables (16x16x4 f32):
//   A  (16x4):  lane (m = lane&15, hi = lane>>4); a[i] = A[m][k0 + 2*hi + i]
//   B  (4x16):  b[i] = B[k0 + 2*hi + i][n = lane&15]
//   C/D (16x16): c[r] = C[m = r + 8*hi][n = lane&15]
__global__ __launch_bounds__(32) void qkv0_wmma(
    const float* __restrict__ X0,
    const float* __restrict__ Wq, const float* __restrict__ bq,
    const float* __restrict__ Wk, const float* __restrict__ bk,
    const float* __restrict__ Wv, const float* __restrict__ bv,
    float* __restrict__ Q0, float* __restrict__ K0, float* __restrict__ V0)
{
  int nt = blockIdx.x, mt = blockIdx.y, z = blockIdx.z;
  const float* W    = (z == 0) ? Wq : (z == 1) ? Wk : Wv;
  const float* bias = (z == 0) ? bq : (z == 1) ? bk : bv;
  float*       out  = (z == 0) ? Q0 : (z == 1) ? K0 : V0;

  int lane = threadIdx.x;
  int m16  = lane & 15;
  int hi   = lane >> 4;
  int arow = (mt * 16 + m16) * DM;
  int ncol = nt * 16 + m16;

  v8f c = {};
  for (int kk = 0; kk < DM; kk += 4) {
    int k0 = kk + 2 * hi;
    v2f a, bm;
    a.x  = X0[arow + k0];
    a.y  = X0[arow + k0 + 1];
    bm.x = W[k0 * DM + ncol];
    bm.y = W[(k0 + 1) * DM + ncol];
    c = __builtin_amdgcn_wmma_f32_16x16x4_f32(false, a, false, bm,
                                              (short)0, c, false, false);
  }

  float bb = bias[ncol];
  bool feat = (z < 2);   // elu(x)+1 feature map for Q, K
  #pragma unroll
  for (int r = 0; r < 8; ++r) {
    int mrow = mt * 16 + r + 8 * hi;
    float v = c[r] + bb;
    if (feat) v = (v > 0.0f) ? (v + 1.0f) : expf(v);
    out[mrow * DM + ncol] = v;
  }
}

// ---------------- scan kernel: one workgroup per batch sample ---------------
__device__ __forceinline__ float block_sum(float v, float* red) {
  int tid = threadIdx.x;
  red[tid] = v;
  __syncthreads();
  if (tid < 256) red[tid] += red[tid + 256] + red[tid + 512];
  __syncthreads();
  for (int s = 128; s > 0; s >>= 1) {
    if (tid < s) red[tid] += red[tid + s];
    __syncthreads();
  }
  float r = red[0];
  __syncthreads();
  return r;
}

__device__ __forceinline__ float gemv_col(const float* __restrict__ W,
                                          const float* h, int n, int K, int N) {
  float acc = 0.0f;
  #pragma unroll 8
  for (int k = 0; k < K; ++k) acc = fmaf(h[k], W[k * N + n], acc);
  return acc;
}

__global__ __launch_bounds__(768) void scan_kernel(
    const float* __restrict__ X0, const float* __restrict__ Q0,
    const float* __restrict__ K0, const float* __restrict__ V0,
    const float* __restrict__ Wq, const float* __restrict__ bq,
    const float* __restrict__ Wk, const float* __restrict__ bk,
    const float* __restrict__ Wv, const float* __restrict__ bv,
    const float* __restrict__ Wo, const float* __restrict__ bo,
    const float* __restrict__ W1, const float* __restrict__ b1,
    const float* __restrict__ W2, const float* __restrict__ b2,
    const float* __restrict__ ln1g, const float* __restrict__ ln1b,
    const float* __restrict__ ln2g, const float* __restrict__ ln2b,
    const float* __restrict__ lnfg, const float* __restrict__ lnfb,
    const int* __restrict__ act_t,
    float* __restrict__ Sbuf, float* __restrict__ Zbuf,
    float* __restrict__ out)
{
  int b = blockIdx.x, tid = threadIdx.x;
  __shared__ float hbuf[DM], qv[DM], kv[DM], vv[DM], attb[DM], h1b[DM];
  __shared__ float ff[FFD], red[DM];
  __shared__ float invh[NH];

  int hh = tid / DH;     // head index
  int m  = tid % DH;     // column within head state

  for (int t = 0; t < TT; ++t) {
    int row = t * BS + b;
    hbuf[tid] = X0[row * DM + tid];
    __syncthreads();

    for (int l = 0; l < NL; ++l) {
      // ---- Q/K/V (with elu+1 feature map on Q,K) ----
      if (l == 0) {
        qv[tid] = Q0[row * DM + tid];
        kv[tid] = K0[row * DM + tid];
        vv[tid] = V0[row * DM + tid];
      } else {
        float q = gemv_col(Wq + l * DM * DM, hbuf, tid, DM, DM) + bq[l * DM + tid];
        float k = gemv_col(Wk + l * DM * DM, hbuf, tid, DM, DM) + bk[l * DM + tid];
        float v = gemv_col(Wv + l * DM * DM, hbuf, tid, DM, DM) + bv[l * DM + tid];
        qv[tid] = (q > 0.0f) ? (q + 1.0f) : expf(q);
        kv[tid] = (k > 0.0f) ? (k + 1.0f) : expf(k);
        vv[tid] = v;
      }
      __syncthreads();

      // ---- Z update and per-head denominator 1/(Q.Z + eps) ----
      float* Zp = Zbuf + (l * BS + b) * DM;
      float zn = Zp[tid] + kv[tid];
      Zp[tid] = zn;
      red[tid] = qv[tid] * zn;
      __syncthreads();
      if (tid < NH) {
        float s = 0.0f;
        for (int d = 0; d < DH; ++d) s += red[tid * DH + d];
        invh[tid] = 1.0f / (s + 1e-6f);
      }
      __syncthreads();

      // ---- fused state update S += k (x) v  and  att = (Q . S) * inv ----
      float* Sp = Sbuf + (((l * BS + b) * NH + hh) * (DH * DH));
      float vm  = vv[hh * DH + m];
      float acc = 0.0f;
      #pragma unroll 4
      for (int d = 0; d < DH; ++d) {
        float s = Sp[d * DH + m] + kv[hh * DH + d] * vm;
        Sp[d * DH + m] = s;
        acc = fmaf(qv[hh * DH + d], s, acc);
      }
      attb[tid] = acc * invh[hh];
      __syncthreads();

      // ---- out-proj + residual + LN1 ----
      float o = gemv_col(Wo + l * DM * DM, attb, tid, DM, DM)
                + bo[l * DM + tid] + hbuf[tid];
      float mean = block_sum(o, red) * (1.0f / DM);
      float d0 = o - mean;
      float var = block_sum(d0 * d0, red) * (1.0f / DM);
      float h1 = d0 * rsqrtf(var + 1e-5f) * ln1g[l * DM + tid] + ln1b[l * DM + tid];
      h1b[tid] = h1;
      __syncthreads();

      // ---- FFN ----
      for (int n = tid; n < FFD; n += DM) {
        float a1 = gemv_col(W1 + l * DM * FFD, h1b, n, DM, FFD) + b1[l * FFD + n];
        ff[n] = fmaxf(a1, 0.0f);
      }
      __syncthreads();
      float f2 = gemv_col(W2 + l * FFD * DM, ff, tid, FFD, DM)
                 + b2[l * DM + tid] + h1;
      float mean2 = block_sum(f2, red) * (1.0f / DM);
      float d2 = f2 - mean2;
      float var2 = block_sum(d2 * d2, red) * (1.0f / DM);
      hbuf[tid] = d2 * rsqrtf(var2 + 1e-5f) * ln2g[l * DM + tid] + ln2b[l * DM + tid];
      __syncthreads();
    }

    // ---- final encoder LN (per-step output) ----
    float xf = hbuf[tid];
    float meanf = block_sum(xf, red) * (1.0f / DM);
    float df = xf - meanf;
    float varf = block_sum(df * df, red) * (1.0f / DM);
    attb[tid] = df * rsqrtf(varf + 1e-5f) * lnfg[tid] + lnfb[tid];
    __syncthreads();

    // ---- gather mu / logvar rows selected by act_tstamps ----
    if (tid < LATENT) {
      for (int a = 0; a < AA; ++a) {
        int ts = act_t[b * AA + a] - 1;
        if (ts < 0) ts = 0;
        if (ts == t) {
          out[(b * AA + a) * LATENT + tid] = attb[tid];                      // mu
          out[BS * AA * LATENT + (b * AA + a) * LATENT + tid] = attb[LATENT + tid]; // logvar
        }
      }
    }
    __syncthreads();
  }
}

// ---------------- host-side launcher ----------------
extern "C" void kernel_launch(void* const* d_in, const int* in_sizes, int n_in,
                              void* d_out, int out_size, void* d_ws, size_t ws_size,
                              hipStream_t stream) {
  (void)in_sizes; (void)n_in; (void)out_size; (void)ws_size;

  const float* x       = (const float*)d_in[0];
  const int*   y       = (const int*)  d_in[1];
  const int*   ind_map = (const int*)  d_in[2];
  const int*   act_t   = (const int*)  d_in[3];
  const float* skel_W  = (const float*)d_in[4];
  const float* skel_b  = (const float*)d_in[5];
  const float* muQ     = (const float*)d_in[6];
  const float* sigQ    = (const float*)d_in[7];
  const float* Wq = (const float*)d_in[8];  const float* bq = (const float*)d_in[9];
  const float* Wk = (const float*)d_in[10]; const float* bk = (const float*)d_in[11];
  const float* Wv = (const float*)d_in[12]; const float* bv = (const float*)d_in[13];
  const float* Wo = (const float*)d_in[14]; const float* bo = (const float*)d_in[15];
  const float* W1 = (const float*)d_in[16]; const float* b1 = (const float*)d_in[17];
  const float* W2 = (const float*)d_in[18]; const float* b2 = (const float*)d_in[19];
  const float* ln1g = (const float*)d_in[20]; const float* ln1b = (const float*)d_in[21];
  const float* ln2g = (const float*)d_in[22]; const float* ln2b = (const float*)d_in[23];
  const float* lnfg = (const float*)d_in[24]; const float* lnfb = (const float*)d_in[25];

  float* ws  = (float*)d_ws;
  float* X0  = ws + OFF_X0;
  float* Q0  = ws + OFF_Q0;
  float* K0b = ws + OFF_K0;
  float* V0  = ws + OFF_V0;
  float* Sb  = ws + OFF_S;
  float* Zb  = ws + OFF_Z;
  float* out = (float*)d_out;

  // 1) zero recurrent state (S, Z) — required every call (graph replay)
  zero_f32<<<1024, 256, 0, stream>>>(Sb, SZ_S + SZ_Z);

  // 2) build layer-0 input rows for all timesteps
  build_x0<<<dim3(TT, BS), 256, 0, stream>>>(x, y, ind_map, skel_W, skel_b,
                                             muQ, sigQ, X0);

  // 3) batched layer-0 Q/K/V via fp32 WMMA (640x768 @ 768x768, x3)
  qkv0_wmma<<<dim3(DM / 16, NR / 16, 3), 32, 0, stream>>>(
      X0, Wq, bq, Wk, bk, Wv, bv, Q0, K0b, V0);

  // 4) recurrent scan: one WG per sample, zero inter-WG sync
  scan_kernel<<<BS, DM, 0, stream>>>(
      X0, Q0, K0b, V0,
      Wq, bq, Wk, bk, Wv, bv, Wo, bo, W1, b1, W2, b2,
      ln1g, ln1b, ln2g, ln2b, lnfg, lnfb,
      act_t, Sb, Zb, out);
}